// SupervisedContrastiveLoss_48533130445659
// MI455X (gfx1250) — compile-verified
//
#include <hip/hip_runtime.h>
#include <math.h>

typedef __attribute__((ext_vector_type(2))) float v2f;
typedef __attribute__((ext_vector_type(8))) float v8f;

#define B_ANCH     1024
#define DIM        128
#define P_POS      8
#define K_NEG      64
#define SLOTS      72            // P + K
#define SLOTS_PAD  80            // padded to 5 tiles of 16
#define NTILES     5
#define LDS_STRIDE 132           // 128 + 4 pad -> conflict-free b64 reads
#define TEMP_INV   14.2857142857142857f   // 1 / 0.07

// One wave (32 lanes) per anchor.
// 1) load + L2-normalize anchor into LDS
// 2) for each of 5 tiles: gather 16 embedding rows (coalesced b128 per row),
//    compute per-row inv-norm via wave reduction, then 32 chained
//    v_wmma_f32_16x16x4_f32 steps: A = anchor replicated over 16 rows,
//    B = 16 normalized embeddings -> row 0 of D = the 16 similarities.
// 3) in-wave LSE over negatives + logaddexp per positive, sum -> ws[b].
__global__ __launch_bounds__(32)
void scl_sim_kernel(const float* __restrict__ anchors,
                    const int*   __restrict__ pos_idx,
                    const int*   __restrict__ neg_idx,
                    const float* __restrict__ all_emb,
                    float*       __restrict__ loss_per_anchor)
{
    __shared__ float ancLDS[DIM];
    __shared__ float tileLDS[16 * LDS_STRIDE];
    __shared__ float simsLDS[SLOTS_PAD];

    const int b    = blockIdx.x;
    const int lane = threadIdx.x;     // 0..31
    const int hi   = lane >> 4;       // 0: K pair {0,1}, 1: K pair {2,3}
    const int n    = lane & 15;       // column / row-in-tile this lane feeds

    // ---- anchor: load, normalize, stash in LDS ----
    {
        const float4* arow = (const float4*)(anchors + (size_t)b * DIM);
        float4 av = arow[lane];       // floats [4*lane .. 4*lane+3]
        float ss = av.x * av.x + av.y * av.y + av.z * av.z + av.w * av.w;
        #pragma unroll
        for (int m = 16; m >= 1; m >>= 1) ss += __shfl_xor(ss, m, 32);
        float inv = 1.0f / fmaxf(sqrtf(ss), 1e-12f);
        float4 an;
        an.x = av.x * inv; an.y = av.y * inv; an.z = av.z * inv; an.w = av.w * inv;
        ((float4*)ancLDS)[lane] = an;
    }
    __syncthreads();

    float row_inv = 1.0f;             // inv-norm of embedding row `n` of current tile

    for (int t = 0; t < NTILES; ++t) {
        // ---- gather 16 rows into LDS; per-row inv-norm to the 2 lanes with n==r
        #pragma unroll
        for (int r = 0; r < 16; ++r) {
            int slot = t * 16 + r;
            int idx;
            if (slot < P_POS)      idx = pos_idx[b * P_POS + slot];
            else if (slot < SLOTS) idx = neg_idx[b * K_NEG + (slot - P_POS)];
            else                   idx = 0;   // pad row, masked out later
            const float4* erow = (const float4*)(all_emb + (size_t)idx * DIM);
            float4 ev = erow[lane];
            ((float4*)(tileLDS + r * LDS_STRIDE))[lane] = ev;
            float ps = ev.x * ev.x + ev.y * ev.y + ev.z * ev.z + ev.w * ev.w;
            #pragma unroll
            for (int m = 16; m >= 1; m >>= 1) ps += __shfl_xor(ps, m, 32);
            if (n == r) row_inv = 1.0f / fmaxf(sqrtf(ps), 1e-12f);
        }
        __syncthreads();

        // ---- 16x16 tile of dots via chained f32 WMMA, K = 128 in steps of 4
        v8f acc = {};
        #pragma unroll
        for (int kc = 0; kc < 32; ++kc) {
            const int c = 4 * kc + 2 * hi;
            // A matrix 16x4: lanes 0-15 hold K={0,1}, lanes 16-31 hold K={2,3};
            // all 16 rows identical == anchor chunk -> plain lane-dependent float2.
            v2f aa = *(const v2f*)(ancLDS + c);
            // B matrix 4x16: lane n (and n+16) feeds column n with the same K split.
            v2f bb = *(const v2f*)(tileLDS + n * LDS_STRIDE + c);
            bb.x *= row_inv;
            bb.y *= row_inv;
            acc = __builtin_amdgcn_wmma_f32_16x16x4_f32(
                false, aa, false, bb, (short)0, acc, false, false);
        }
        // D row M=0 lives in acc VGPR0, lanes 0-15 (lanes 16-31 duplicate row 8).
        if (lane < 16) simsLDS[t * 16 + lane] = acc[0] * TEMP_INV;
        __syncthreads();
    }

    // ---- per-anchor loss: stable LSE over 64 negatives (slots 8..71)
    float a0 = simsLDS[P_POS + lane];
    float a1 = simsLDS[P_POS + 32 + lane];
    float mx = fmaxf(a0, a1);
    #pragma unroll
    for (int m = 16; m >= 1; m >>= 1) mx = fmaxf(mx, __shfl_xor(mx, m, 32));
    float e = expf(a0 - mx) + expf(a1 - mx);
    #pragma unroll
    for (int m = 16; m >= 1; m >>= 1) e += __shfl_xor(e, m, 32);
    float L = mx + logf(e);

    // positives: loss_p = -p + logaddexp(p, L)
    float lsum = 0.0f;
    if (lane < P_POS) {
        float p  = simsLDS[lane];
        float mm = fmaxf(p, L);
        lsum = -p + mm + logf(expf(p - mm) + expf(L - mm));
    }
    #pragma unroll
    for (int m = 16; m >= 1; m >>= 1) lsum += __shfl_xor(lsum, m, 32);
    if (lane == 0) loss_per_anchor[b] = lsum;
}

// Deterministic fixed-order reduction of the 1024 per-anchor sums -> mean.
__global__ __launch_bounds__(256)
void scl_reduce_kernel(const float* __restrict__ loss_per_anchor,
                       float* __restrict__ out)
{
    __shared__ float red[256];
    const int t = threadIdx.x;
    float s = 0.0f;
    #pragma unroll
    for (int i = 0; i < B_ANCH / 256; ++i) s += loss_per_anchor[t + i * 256];
    red[t] = s;
    __syncthreads();
    #pragma unroll
    for (int off = 128; off >= 1; off >>= 1) {
        if (t < off) red[t] += red[t + off];
        __syncthreads();
    }
    if (t == 0) out[0] = red[0] * (1.0f / (float)(B_ANCH * P_POS));
}

extern "C" void kernel_launch(void* const* d_in, const int* in_sizes, int n_in,
                              void* d_out, int out_size, void* d_ws, size_t ws_size,
                              hipStream_t stream)
{
    const float* anchors = (const float*)d_in[0];   // [1024,128] f32
    const int*   pos_idx = (const int*)  d_in[1];   // [1024,8]   i32
    const int*   neg_idx = (const int*)  d_in[2];   // [1024,64]  i32
    const float* all_emb = (const float*)d_in[3];   // [200000,128] f32
    float* loss_pa = (float*)d_ws;                  // 1024 floats of scratch

    scl_sim_kernel<<<dim3(B_ANCH), dim3(32), 0, stream>>>(
        anchors, pos_idx, neg_idx, all_emb, loss_pa);
    scl_reduce_kernel<<<dim3(1), dim3(256), 0, stream>>>(
        loss_pa, (float*)d_out);
}